// NewMF_21801253994971
// MI455X (gfx1250) — compile-verified
//
#include <hip/hip_runtime.h>

// NewMF: out[b] = sigmoid( sum_d  prod_{l: items[b,l]!=-1}  F[items[b,l], d] )
// B=16384 rows, L=50 (tail-padded with -1), D=128 factors, table 100000x128 f32.
//
// Traffic-bound gather kernel: ~210MB of embedding reads from a 51MB table
// (fits in MI455X's 192MB L2). One wave32 per row; lane owns 4 factors
// (global_load_b128, one full 512B embedding row per wave request).
// Item ids are wave-uniform -> v_readlane broadcast to SGPR address.
// Valid-count (tail padding) computed once -> dynamic trip count halves traffic.
// Factor-sum done with V_WMMA_F32_16X16X4_F32 (B = ones => column sums),
// which is layout-safe and free relative to memory.

typedef float f4  __attribute__((ext_vector_type(4)));
typedef float v2f __attribute__((ext_vector_type(2)));
typedef float v8f __attribute__((ext_vector_type(8)));

// ds_swizzle group-of-32: offset[15]=0, and_mask=0x1F, xor_mask in [14:10]
#define SWZ_XOR(x, m) __builtin_amdgcn_ds_swizzle((x), (((m) << 10) | 0x1F))

#define ROWS_PER_BLOCK 8   // 8 waves of 32 per block
#define L_MAX 50
#define D_FACTORS 128

__global__ __launch_bounds__(256)
void newmf_prod_reduce_kernel(const int* __restrict__ items,
                              const float* __restrict__ factors,
                              float* __restrict__ out,
                              int nrows)
{
    const int lane = threadIdx.x & 31;
    const int wave = threadIdx.x >> 5;
    const int row  = blockIdx.x * ROWS_PER_BLOCK + wave;
    if (row >= nrows) return;   // wave-uniform; EXEC stays all-ones below

    // ---- load the 50 (wave-uniform) item ids with two coalesced lane loads
    const int* it_row = items + row * L_MAX;
    int ia = it_row[lane];                                   // ids 0..31
    int ib = (lane < (L_MAX - 32)) ? it_row[32 + lane] : -1; // ids 32..49

    // ---- count valid items (padding is a -1 tail): flags + butterfly reduce
    int v = (ia != -1) + (ib != -1);
    v += SWZ_XOR(v, 1);
    v += SWZ_XOR(v, 2);
    v += SWZ_XOR(v, 4);
    v += SWZ_XOR(v, 8);
    v += SWZ_XOR(v, 16);
    const int cnt = __builtin_amdgcn_readfirstlane(v);    // >= 1 by construction

    // ---- gather-product loop: lane owns factors [4*lane, 4*lane+3]
    f4 acc = {1.0f, 1.0f, 1.0f, 1.0f};
    {
        int it0 = __builtin_amdgcn_readlane(ia, 0);
        const f4* p0 = (const f4*)(factors + (long long)it0 * D_FACTORS);
        f4 cur = p0[lane];
        for (int t = 1; t < cnt; ++t) {                   // scalar loop bound
            int src = (t < 32) ? ia : ib;                 // uniform select
            int it  = __builtin_amdgcn_readlane(src, t & 31); // SGPR id
            const f4* p = (const f4*)(factors + (long long)it * D_FACTORS);
            f4 nxt = p[lane];                             // prefetch next row
            acc *= cur;                                   // multiply current
            cur = nxt;
        }
        acc *= cur;
    }

    // ---- sum the 128 per-row values with WMMA (B = ones):
    // D[m,n] = sum_k A[m,k] + C[m,n]  for any B lane-layout, so two chained
    // 16x16x4 f32 WMMAs fold all 4 accumulator components of all 32 lanes
    // into column sums; then 8-component add + xor-16 swizzle gives the total.
    v2f a01 = {acc.x, acc.y};
    v2f a23 = {acc.z, acc.w};
    v2f bones = {1.0f, 1.0f};
    v8f c = {};
    c = __builtin_amdgcn_wmma_f32_16x16x4_f32(false, a01, false, bones,
                                              (short)0, c, false, false);
    c = __builtin_amdgcn_wmma_f32_16x16x4_f32(false, a23, false, bones,
                                              (short)0, c, false, false);
    float s = ((c[0] + c[1]) + (c[2] + c[3])) + ((c[4] + c[5]) + (c[6] + c[7]));
    s += __int_as_float(SWZ_XOR(__float_as_int(s), 16));  // fold M halves

    if (lane == 0) {
        // sigmoid: fast path — v_exp_f32 + v_rcp_f32 (no IEEE div sequence)
        out[row] = __builtin_amdgcn_rcpf(1.0f + __expf(-s));
    }
}

extern "C" void kernel_launch(void* const* d_in, const int* in_sizes, int n_in,
                              void* d_out, int out_size, void* d_ws, size_t ws_size,
                              hipStream_t stream) {
    (void)n_in; (void)d_ws; (void)ws_size;
    const int*   items   = (const int*)d_in[0];     // [B, 50] int32
    const float* factors = (const float*)d_in[1];   // [100000, 128] f32
    float*       out     = (float*)d_out;           // [B] f32

    const int nrows = out_size;                     // B = 16384
    const int blocks = (nrows + ROWS_PER_BLOCK - 1) / ROWS_PER_BLOCK;
    newmf_prod_reduce_kernel<<<blocks, 256, 0, stream>>>(items, factors, out, nrows);
}